// CandidateIndex_90907277787717
// MI455X (gfx1250) — compile-verified
//
#include <hip/hip_runtime.h>
#include <stdint.h>

// Problem constants (match reference)
#define Bq   128
#define Dd   64
#define Xn   1000000
#define Kk   100
#define NN0  32
#define KP   132          // K + N0
#define NB   1024         // histogram bins (top 10 bits of ordered float)
#define CAP  4096         // candidate capacity per row (power of two)
#define NCH  128          // column chunks
#define CHUNK 7840        // 490 tiles of 16 cols; 128*7840 = 1,003,520 >= X
#define TILES_PER_CHUNK 490

typedef __attribute__((ext_vector_type(2))) float v2f;
typedef __attribute__((ext_vector_type(8))) float v8f;

// Monotone float <-> uint32 order-preserving map
__device__ __forceinline__ uint32_t fkey(float f) {
    uint32_t u = __float_as_uint(f);
    return (u & 0x80000000u) ? ~u : (u | 0x80000000u);
}
__device__ __forceinline__ float finv(uint32_t m) {
    uint32_t u = (m & 0x80000000u) ? (m & 0x7fffffffu) : ~m;
    return __uint_as_float(u);
}

// One kernel, two modes:
//   mode 0: GEMM + per-row LDS histogram -> global histogram
//   mode 1: GEMM + collect candidates with score >= thresh[row]
__global__ __launch_bounds__(256)
void gemm_pass(const float* __restrict__ Q, const float* __restrict__ E,
               uint32_t* __restrict__ ghist, const float* __restrict__ thresh,
               uint32_t* __restrict__ cnt, unsigned long long* __restrict__ cand,
               int mode)
{
    __shared__ uint32_t lhist[16 * NB];   // 64 KB

    const int tid  = threadIdx.x;
    const int lane = tid & 31;
    const int wave = tid >> 5;
    const int row0 = blockIdx.x * 16;              // row group (x fastest -> L2 chunk reuse)
    const int chunk0 = blockIdx.y * CHUNK;

    if (mode == 0) {
        for (int i = tid; i < 16 * NB; i += 256) lhist[i] = 0;
        __syncthreads();
    }

    const int m  = lane & 15;     // M (A) / N (B,C)
    const int kp = lane >> 4;     // K-pair selector / C row-half selector

    // A fragments: 16x4 f32 tiles covering K=0..63
    // (lanes 0-15 hold K{k0,k0+1}; lanes 16-31 hold K{k0+2,k0+3})
    v2f a[16];
#pragma unroll
    for (int kk = 0; kk < 16; ++kk) {
        int k0 = kk * 4 + 2 * kp;
        a[kk].x = Q[(row0 + m) * Dd + k0];
        a[kk].y = Q[(row0 + m) * Dd + k0 + 1];
    }

    float th[8];
    if (mode == 1) {
#pragma unroll
        for (int r = 0; r < 8; ++r) th[r] = thresh[row0 + r + 8 * kp];
    }

    const size_t sX = (size_t)Xn;

    for (int t = wave; t < TILES_PER_CHUNK; t += 8) {
        const int tile0 = chunk0 + t * 16;
        const int col   = tile0 + m;
        const bool ok   = (col < Xn);
        const int colc  = ok ? col : (Xn - 1);   // clamped: loads always in-bounds

        // Batch all B-fragment loads (no predication -> no exec-mask branches,
        // single loadcnt wait before the WMMA chain)
        v2f b[16];
#pragma unroll
        for (int kk = 0; kk < 16; ++kk) {
            const float* p = E + (size_t)(kk * 4 + 2 * kp) * sX + colc;
            b[kk].x = p[0];
            b[kk].y = p[sX];
        }

        // Two independent accumulation chains over K
        v8f c0 = {}, c1 = {};
#pragma unroll
        for (int kk = 0; kk < 16; kk += 2) {
            c0 = __builtin_amdgcn_wmma_f32_16x16x4_f32(false, a[kk],     false, b[kk],
                                                       (short)0, c0, false, false);
            c1 = __builtin_amdgcn_wmma_f32_16x16x4_f32(false, a[kk + 1], false, b[kk + 1],
                                                       (short)0, c1, false, false);
        }
        v8f c = c0 + c1;

        if (ok) {
            union { v8f v; float f[8]; } u; u.v = c;
#pragma unroll
            for (int r = 0; r < 8; ++r) {
                const float s  = u.f[r];
                const int   rl = r + 8 * kp;   // local row 0..15
                if (mode == 0) {
                    atomicAdd(&lhist[rl * NB + (fkey(s) >> 22)], 1u);
                } else {
                    if (s >= th[r]) {
                        const int row = row0 + rl;
                        uint32_t pos = atomicAdd(&cnt[row], 1u);
                        if (pos < CAP) {
                            unsigned long long key =
                                ((unsigned long long)fkey(s) << 32) |
                                (unsigned long long)(uint32_t)(~(uint32_t)col);
                            cand[(size_t)row * CAP + pos] = key;
                        }
                    }
                }
            }
        }
    }

    if (mode == 0) {
        __syncthreads();
        for (int i = tid; i < 16 * NB; i += 256) {
            uint32_t v = lhist[i];
            if (v) atomicAdd(&ghist[row0 * NB + i], v);
        }
    }
}

// Per-row threshold = lower edge of the bin where cumulative-from-top reaches KP
__global__ void thresh_kernel(const uint32_t* __restrict__ ghist,
                              float* __restrict__ thresh)
{
    int row = threadIdx.x;
    if (row < Bq) {
        uint32_t cum = 0;
        int binT = 0;
        for (int b = NB - 1; b >= 0; --b) {
            cum += ghist[row * NB + b];
            if (cum >= KP) { binT = b; break; }
        }
        thresh[row] = finv((uint32_t)binT << 22);
    }
}

// One block per row: bitonic-sort candidates, mask invalid ids, emit first K valid
__global__ __launch_bounds__(256)
void finalize_kernel(const uint32_t* __restrict__ cnt,
                     const unsigned long long* __restrict__ cand,
                     const int* __restrict__ item_ids,
                     const int* __restrict__ invalid_ids,
                     float* __restrict__ out)
{
    __shared__ unsigned long long keys[CAP];   // 32 KB
    __shared__ int   inv[NN0];
    __shared__ float ssc[KP];
    __shared__ int   sid[KP];
    __shared__ int   sval[KP];

    const int row = blockIdx.x;
    const int tid = threadIdx.x;

    uint32_t n = cnt[row];
    if (n > CAP) n = CAP;
    for (int i = tid; i < CAP; i += 256)
        keys[i] = (i < (int)n) ? cand[(size_t)row * CAP + i] : 0ull;
    if (tid < NN0) inv[tid] = invalid_ids[row * NN0 + tid];
    __syncthreads();

    // bitonic sort ascending; result read from the high end
    for (int k2 = 2; k2 <= CAP; k2 <<= 1) {
        for (int j = k2 >> 1; j > 0; j >>= 1) {
            for (int i = tid; i < CAP; i += 256) {
                int ixj = i ^ j;
                if (ixj > i) {
                    unsigned long long x = keys[i], y = keys[ixj];
                    bool asc = ((i & k2) == 0);
                    if ((x > y) == asc) { keys[i] = y; keys[ixj] = x; }
                }
            }
            __syncthreads();
        }
    }

    if (tid < KP) {
        unsigned long long key = keys[CAP - 1 - tid];
        uint32_t colc = ~(uint32_t)key;               // low 32 bits were ~col
        float    sc   = finv((uint32_t)(key >> 32));
        int      id   = (colc < (uint32_t)Xn) ? item_ids[colc] : 0;
        bool valid = (key != 0ull);
#pragma unroll
        for (int q = 0; q < NN0; ++q) valid = valid && (id != inv[q]);
        ssc[tid] = sc; sid[tid] = id; sval[tid] = valid ? 1 : 0;
    }
    __syncthreads();

    if (tid == 0) {
        int outc = 0;
        for (int j = 0; j < KP && outc < Kk; ++j) {
            if (sval[j]) {
                out[row * Kk + outc]           = (float)sid[j];
                out[Bq * Kk + row * Kk + outc] = ssc[j];
                ++outc;
            }
        }
        for (; outc < Kk; ++outc) {
            out[row * Kk + outc] = 0.0f;
            out[Bq * Kk + row * Kk + outc] = 0.0f;
        }
    }
}

extern "C" void kernel_launch(void* const* d_in, const int* in_sizes, int n_in,
                              void* d_out, int out_size, void* d_ws, size_t ws_size,
                              hipStream_t stream)
{
    (void)in_sizes; (void)n_in; (void)out_size; (void)ws_size;

    const float* Q  = (const float*)d_in[0];          // (128,64)
    const float* E  = (const float*)d_in[1];          // (64,1e6) row-major
    const int* item_ids    = (const int*)d_in[2];     // (1e6)
    const int* invalid_ids = (const int*)d_in[3];     // (128,32)

    // workspace layout
    uint8_t* w = (uint8_t*)d_ws;
    uint32_t* ghist = (uint32_t*)(w);                               // 128*1024*4 = 524288 B
    uint32_t* cnt   = (uint32_t*)(w + 524288);                      // 512 B
    float*    thr   = (float*)   (w + 524288 + 512);                // 512 B
    unsigned long long* cand = (unsigned long long*)(w + 525312);   // 128*4096*8 = 4 MB

    // zero histogram + candidate counters
    hipMemsetAsync(d_ws, 0, 524288 + 512, stream);

    dim3 grid(8, NCH);   // x = row-group (fastest -> L2 reuse of E chunk across the 8 consumers)
    gemm_pass<<<grid, 256, 0, stream>>>(Q, E, ghist, thr, cnt, cand, 0);
    thresh_kernel<<<1, 128, 0, stream>>>(ghist, thr);
    gemm_pass<<<grid, 256, 0, stream>>>(Q, E, ghist, thr, cnt, cand, 1);
    finalize_kernel<<<Bq, 256, 0, stream>>>(cnt, cand, item_ids, invalid_ids, (float*)d_out);
}